// NodeUpdate_82781199663586
// MI455X (gfx1250) — compile-verified
//
#include <hip/hip_runtime.h>

typedef float v2f __attribute__((ext_vector_type(2)));
typedef float v8f __attribute__((ext_vector_type(8)));
typedef unsigned int u32x4 __attribute__((ext_vector_type(4)));
typedef int i32x4 __attribute__((ext_vector_type(4)));
typedef int i32x8 __attribute__((ext_vector_type(8)));

#define N_NODES 10000
#define DIM 128
#define BN_EPS 1e-5f

// ---------------------------------------------------------------------------
// K1: agg = x  (folds the "(1+eps)*x_i" term, eps=0, into the accumulator
// init so the scatter kernel only adds). Also zeroes the BN stat accumulators.
// ---------------------------------------------------------------------------
__global__ void k_init(const float4* __restrict__ x4, float4* __restrict__ agg4,
                       float* __restrict__ colsum, float* __restrict__ colsumsq,
                       int n4) {
    int i = blockIdx.x * blockDim.x + threadIdx.x;
    if (i < n4) agg4[i] = x4[i];
    if (blockIdx.x == 0) {
        if (threadIdx.x < DIM)               colsum[threadIdx.x]         = 0.0f;
        else if (threadIdx.x < 2 * DIM)      colsumsq[threadIdx.x - DIM] = 0.0f;
    }
}

// ---------------------------------------------------------------------------
// K2: edge scatter-add. One wave (32 lanes) per edge; each lane moves a
// float4 (coalesced 512B read of x[src]) and issues 4 hardware f32 atomics
// into agg[dst]. x and agg are L2-resident (5.1 MB each vs 192 MB L2), so
// this roofline-dominant phase (~655 MB of traffic) runs at L2 speed.
// ---------------------------------------------------------------------------
__global__ __launch_bounds__(256) void k_scatter(const float4* __restrict__ x4,
                                                 const int* __restrict__ src,
                                                 const int* __restrict__ dst,
                                                 float* __restrict__ agg,
                                                 int nEdges) {
    int t    = blockIdx.x * blockDim.x + threadIdx.x;
    int e    = t >> 5;
    int lane = t & 31;
    if (e >= nEdges) return;
    int s = src[e];
    int d = dst[e];
    float4 v = x4[s * (DIM / 4) + lane];
    float* p = agg + (size_t)d * DIM + lane * 4;
    unsafeAtomicAdd(p + 0, v.x);
    unsafeAtomicAdd(p + 1, v.y);
    unsafeAtomicAdd(p + 2, v.z);
    unsafeAtomicAdd(p + 3, v.w);
}

// ---------------------------------------------------------------------------
// K3: h = relu(agg @ W + b) via V_WMMA_F32_16X16X4_F32, fused with per-column
// sum / sum-of-squares accumulation for BatchNorm.
//   grid.x = 625 M-tiles; block = 256 threads = 8 waves; wave w owns N-tile w.
//   A tile (16x128 f32, 8 KB contiguous) is DMA'd into LDS by the Tensor
//   Data Mover; D# pad fields insert 4 DWORDs every 128 DWORDs, giving the
//   bank-conflict-free 132-float row stride in hardware.
// A-frag layout (32-bit A 16x4): lanes 0-15: M=lane, V0=K+0, V1=K+1;
//                                lanes 16-31: M=lane-16, V0=K+2, V1=K+3.
// C layout: VGPR i -> M = i + (lane<16 ? 0 : 8), N = lane&15.
// ---------------------------------------------------------------------------
__global__ __launch_bounds__(256) void k_gemm(const float* __restrict__ agg,
                                              const float* __restrict__ W,
                                              const float* __restrict__ bias,
                                              float* __restrict__ h,
                                              float* __restrict__ colsum,
                                              float* __restrict__ colsumsq) {
    __shared__ float As[16][132];

    const int tid   = threadIdx.x;
    const int tileM = blockIdx.x;                 // 0..624

#if __has_builtin(__builtin_amdgcn_tensor_load_to_lds) && __has_builtin(__builtin_amdgcn_s_wait_tensorcnt)
    // --- TDM path: one DMA descriptor moves the whole A tile into LDS. ---
    if (tid < 32) {                               // wave 0 issues the TDM op
        unsigned int lds_off = (unsigned int)(unsigned long long)(&As[0][0]);
        unsigned long long gaddr =
            (unsigned long long)(const void*)(agg + (size_t)tileM * 16 * DIM);

        u32x4 g0;
        g0[0] = 1u;                               // count=1, user descriptor
        g0[1] = lds_off;                          // lds_addr (bytes)
        g0[2] = (unsigned int)gaddr;              // global_addr[31:0]
        g0[3] = ((unsigned int)(gaddr >> 32) & 0x1FFFFFFu)  // global_addr[56:32]
                | (2u << 30);                     // type = 2 ("image")

        i32x8 g1;
        g1[0] = (2 << 16)                         // data_size = 4B
              | (1 << 20)                         // pad_enable
              | (6 << 22)                         // pad_interval: every 128 DW
              | (3 << 25);                        // pad_amount: 4 DW (128->132)
        g1[1] = (int)(2048u << 16);               // tensor_dim0[15:0]=2048 in [31:16]
        g1[2] = (1 << 16);                        // tensor_dim0[31:16]=0; tensor_dim1=1
        g1[3] = (int)(2048u << 16);               // tile_dim0 = 2048 elements
        g1[4] = 0;                                // tile_dim1=0, tile_dim2=0 (1D)
        g1[5] = 2048;                             // tensor_dim0_stride[31:0]
        g1[6] = 0;
        g1[7] = 0;

        i32x4 gz4 = {0, 0, 0, 0};                 // groups 2/3 unused (<=2D)
        i32x8 gz8 = {0, 0, 0, 0, 0, 0, 0, 0};     // extra group (clang-23 form)
        __builtin_amdgcn_tensor_load_to_lds(g0, g1, gz4, gz4, gz8, 0);
        __builtin_amdgcn_s_wait_tensorcnt((short)0);
    }
#else
    // --- Fallback: cooperative coalesced copy (512 float4, 2/thread). ---
    {
        const float4* g = reinterpret_cast<const float4*>(agg + (size_t)tileM * 16 * DIM);
        #pragma unroll
        for (int it = 0; it < 2; ++it) {
            int idx = tid + it * 256;
            int r   = idx >> 5;
            int c4  = idx & 31;
            float4 v = g[r * 32 + c4];
            As[r][c4 * 4 + 0] = v.x;
            As[r][c4 * 4 + 1] = v.y;
            As[r][c4 * 4 + 2] = v.z;
            As[r][c4 * 4 + 3] = v.w;
        }
    }
#endif
    __syncthreads();

    const int wave  = tid >> 5;                   // N-tile 0..7
    const int lane  = tid & 31;
    const int nl    = lane & 15;
    const int koff  = (lane < 16) ? 0 : 2;
    const int nBase = wave * 16;
    const float* Wp = W + nBase + nl;             // column n, row k -> Wp[k*DIM]

    v8f c = {};
    #pragma unroll 8
    for (int kk = 0; kk < DIM; kk += 4) {
        v2f a;
        a.x = As[nl][kk + koff];
        a.y = As[nl][kk + koff + 1];
        v2f b;
        b.x = Wp[(size_t)(kk + koff) * DIM];
        b.y = Wp[(size_t)(kk + koff + 1) * DIM];
        c = __builtin_amdgcn_wmma_f32_16x16x4_f32(
                /*neg_a=*/false, a, /*neg_b=*/false, b,
                /*c_mod=*/(short)0, c, /*reuse_a=*/false, /*reuse_b=*/false);
    }

    // Epilogue: bias + ReLU + store + per-column BN partial stats.
    const float bn      = bias[nBase + nl];
    const int   rowBase = tileM * 16 + ((lane < 16) ? 0 : 8);
    float s = 0.0f, sq = 0.0f;
    #pragma unroll
    for (int i = 0; i < 8; ++i) {
        float v = c[i] + bn;
        v = fmaxf(v, 0.0f);
        h[(size_t)(rowBase + i) * DIM + nBase + nl] = v;
        s  += v;
        sq += v * v;
    }
    // Lanes L and L+16 cover rows 0-7 / 8-15 of the same column: 2 atomics/col.
    unsafeAtomicAdd(&colsum[nBase + nl],   s);
    unsafeAtomicAdd(&colsumsq[nBase + nl], sq);
}

// ---------------------------------------------------------------------------
// K4: fold BN stats into per-column scale/shift.
// ---------------------------------------------------------------------------
__global__ void k_stats(const float* __restrict__ colsum,
                        const float* __restrict__ colsumsq,
                        const float* __restrict__ gamma,
                        const float* __restrict__ beta,
                        float* __restrict__ scale,
                        float* __restrict__ shift) {
    int t = threadIdx.x;
    if (t < DIM) {
        const float invN = 1.0f / (float)N_NODES;
        float mean = colsum[t] * invN;
        float var  = colsumsq[t] * invN - mean * mean;
        float sc   = gamma[t] * rsqrtf(var + BN_EPS);
        scale[t] = sc;
        shift[t] = beta[t] - mean * sc;
    }
}

// ---------------------------------------------------------------------------
// K5: apply BN in place on d_out.
// ---------------------------------------------------------------------------
__global__ void k_bn(float* __restrict__ h,
                     const float* __restrict__ scale,
                     const float* __restrict__ shift, int n) {
    int i = blockIdx.x * blockDim.x + threadIdx.x;
    if (i < n) {
        int col = i & (DIM - 1);
        h[i] = h[i] * scale[col] + shift[col];
    }
}

// ---------------------------------------------------------------------------
extern "C" void kernel_launch(void* const* d_in, const int* in_sizes, int n_in,
                              void* d_out, int out_size, void* d_ws, size_t ws_size,
                              hipStream_t stream) {
    const float* x     = (const float*)d_in[0];
    const int*   ei    = (const int*)d_in[1];   // [2, E] flat: src then dst
    const float* W     = (const float*)d_in[2];
    const float* bias  = (const float*)d_in[3];
    const float* gamma = (const float*)d_in[4];
    const float* beta  = (const float*)d_in[5];
    float*       out   = (float*)d_out;

    const int nEdges = in_sizes[1] / 2;

    // Workspace layout: agg[N*D] | colsum[D] | colsumsq[D] | scale[D] | shift[D]
    float* agg      = (float*)d_ws;
    float* colsum   = agg + (size_t)N_NODES * DIM;
    float* colsumsq = colsum + DIM;
    float* scale    = colsumsq + DIM;
    float* shift    = scale + DIM;

    // K1: agg = x, zero stats
    const int n4 = N_NODES * DIM / 4;
    k_init<<<(n4 + 255) / 256, 256, 0, stream>>>(
        (const float4*)x, (float4*)agg, colsum, colsumsq, n4);

    // K2: edge scatter-add (one wave per edge)
    {
        long long total = (long long)nEdges * 32;
        int blocks = (int)((total + 255) / 256);
        k_scatter<<<blocks, 256, 0, stream>>>(
            (const float4*)x, ei, ei + nEdges, agg, nEdges);
    }

    // K3: TDM tile stage + WMMA GEMM + bias + ReLU + BN stat atomics
    k_gemm<<<N_NODES / 16, 256, 0, stream>>>(agg, W, bias, out, colsum, colsumsq);

    // K4: BN coefficients
    k_stats<<<1, 128, 0, stream>>>(colsum, colsumsq, gamma, beta, scale, shift);

    // K5: BN apply in place
    const int n = N_NODES * DIM;
    k_bn<<<(n + 255) / 256, 256, 0, stream>>>(out, scale, shift, n);
}